// Expected_Depth_Loss_beta_39719857554148
// MI455X (gfx1250) — compile-verified
//
#include <hip/hip_runtime.h>
#include <cstdint>

#define S_TOT   8192
#define N_NODE  64
#define SW      16
#define N_OPS   8
#define E_TOT   2016          // 64*63/2 node pairs
#define B4      (E_TOT / 4)   // 504 x b128 copies per sample
#define BLK     256

typedef int v4i __attribute__((ext_vector_type(4)));
typedef __attribute__((address_space(1))) v4i as1_v4i;
typedef __attribute__((address_space(3))) v4i as3_v4i;
typedef __attribute__((address_space(3))) void as3_void;

// ---- CDNA5 async global->LDS staging (ASYNCcnt path) -----------------------
__device__ __forceinline__ void async_copy16(const float* gsrc, float* ldst) {
#if defined(__has_builtin) && __has_builtin(__builtin_amdgcn_global_load_async_to_lds_b128)
    __builtin_amdgcn_global_load_async_to_lds_b128((as1_v4i*)gsrc, (as3_v4i*)ldst, 0, 0);
#else
    unsigned l = (unsigned)(uintptr_t)(as3_void*)ldst;
    asm volatile("global_load_async_to_lds_b128 %0, %1, off"
                 :: "v"(l), "v"(gsrc) : "memory");
#endif
}

__device__ __forceinline__ void wait_async0() {
#if defined(__has_builtin) && __has_builtin(__builtin_amdgcn_s_wait_asynccnt)
    __builtin_amdgcn_s_wait_asynccnt(0);
#else
    asm volatile("s_wait_asynccnt 0" ::: "memory");
#endif
}

// ---- deterministic block reductions (wave32-aware) -------------------------
__device__ __forceinline__ float block_max(float v, float* s_red, int tid) {
#pragma unroll
    for (int o = 16; o >= 1; o >>= 1) v = fmaxf(v, __shfl_xor(v, o, 32));
    __syncthreads();
    if ((tid & 31) == 0) s_red[tid >> 5] = v;
    __syncthreads();
    if (tid < 32) {
        float x = (tid < 8) ? s_red[tid] : -3.4e38f;
#pragma unroll
        for (int o = 4; o >= 1; o >>= 1) x = fmaxf(x, __shfl_xor(x, o, 32));
        if (tid == 0) s_red[8] = x;
    }
    __syncthreads();
    return s_red[8];
}

__device__ __forceinline__ float block_sum(float v, float* s_red, int tid) {
#pragma unroll
    for (int o = 16; o >= 1; o >>= 1) v += __shfl_xor(v, o, 32);
    __syncthreads();
    if ((tid & 31) == 0) s_red[tid >> 5] = v;
    __syncthreads();
    if (tid < 32) {
        float x = (tid < 8) ? s_red[tid] : 0.0f;
#pragma unroll
        for (int o = 4; o >= 1; o >>= 1) x += __shfl_xor(x, o, 32);
        if (tid == 0) s_red[8] = x;
    }
    __syncthreads();
    return s_red[8];
}

// ---- kernel 1: one workgroup per sample ------------------------------------
__global__ __launch_bounds__(BLK) void depth_kernel(
    const float* __restrict__ alpha,   // [S,64,16,8]
    const float* __restrict__ beta,    // [S,2016]
    const float* __restrict__ theta,   // [S]
    float* __restrict__ ws)            // [S] partial results
{
    __shared__ __align__(16) float s_beta[E_TOT];
    __shared__ float s_w[N_NODE * SW];
    __shared__ float s_ed[N_NODE + 2];
    __shared__ float s_red[9];

    const int tid = threadIdx.x;
    const int s   = blockIdx.x;

    // (1) kick off async staging of this sample's beta row into LDS
    const float* gb = beta + (size_t)s * E_TOT;
    for (int q = tid; q < B4; q += BLK)
        async_copy16(gb + q * 4, s_beta + q * 4);

    if (tid < N_NODE + 2) s_ed[tid] = 0.0f;

    // (2) edge_max over first 7 ops (coalesced float4 pairs)
    const float* ga = alpha + (size_t)s * (N_NODE * SW * N_OPS);
    for (int p = tid; p < N_NODE * SW; p += BLK) {
        const float4* a = (const float4*)(ga + (size_t)p * N_OPS);
        float4 x = a[0], y = a[1];
        float m = fmaxf(fmaxf(fmaxf(x.x, x.y), fmaxf(x.z, x.w)),
                        fmaxf(y.x, fmaxf(y.y, y.z)));   // exclude op 7 (y.w)
        s_w[p] = m;
    }
    __syncthreads();

    // (3) masked softmax of each node's 16-wide window; invalid slots -> 0
    if (tid < N_NODE) {
        const int jn = tid;
        const int r  = min(jn + 2, SW);
        float* wrow  = s_w + jn * SW;
        float m = -3.4e38f;
        for (int k = 0; k < r; ++k) m = fmaxf(m, wrow[k]);
        float sum = 0.0f;
        for (int k = 0; k < r; ++k) { float e = __expf(wrow[k] - m); wrow[k] = e; sum += e; }
        float inv = 1.0f / sum;
        for (int k = 0; k < SW; ++k) wrow[k] = (k < r) ? wrow[k] * inv : 0.0f;
    }
    __syncthreads();

    // (4) sequential expected-depth DP; wave 0, lanes 0..15 carry window terms
    if (tid < 32) {
        for (int jn = 0; jn < N_NODE; ++jn) {
            const int j = jn + 2;
            const int r = min(j, SW);
            float t = 0.0f;
            if (tid < SW) {
                const int pred = j - r + tid;            // in [0, j-1]
                t = s_w[jn * SW + tid] * (s_ed[pred] + 1.0f);
            }
            t += __shfl_xor(t, 8, 32);
            t += __shfl_xor(t, 4, 32);
            t += __shfl_xor(t, 2, 32);
            t += __shfl_xor(t, 1, 32);
            if (tid == 0) s_ed[j] = t;                   // in-order LDS within wave
        }
    }

    // (5) beta is needed from here on: drain ASYNCcnt, then full barrier
    wait_async0();
    __syncthreads();

    // (6) softmax(beta) dotted with ed[i]+ed[j] over all pairs
    float lmax = -3.4e38f;
    for (int e = tid; e < E_TOT; e += BLK) lmax = fmaxf(lmax, s_beta[e]);
    const float bmax = block_max(lmax, s_red, tid);

    float lsum = 0.0f, ldot = 0.0f;
    for (int e = tid; e < E_TOT; e += BLK) {
        float wgt = __expf(s_beta[e] - bmax);
        lsum += wgt;
        // decode linear pair index e -> row m (i = m+2), col -> j
        int m = (int)((127.0f - sqrtf(16129.0f - 8.0f * (float)e)) * 0.5f);
        while (m * (127 - m) / 2 > e) --m;                 // fixup fp rounding
        while ((m + 1) * (126 - m) / 2 <= e) ++m;
        const int i  = m + 2;
        const int j  = m + 3 + (e - m * (127 - m) / 2);
        ldot += wgt * (s_ed[i] + s_ed[j]);
    }
    const float bsum = block_sum(lsum, s_red, tid);
    const float bdot = block_sum(ldot, s_red, tid);

    if (tid == 0) ws[s] = theta[s] * (bdot / bsum);
}

// ---- kernel 2: deterministic final reduction over samples ------------------
__global__ __launch_bounds__(BLK) void final_reduce_kernel(
    const float* __restrict__ ws, float* __restrict__ out)
{
    __shared__ float s_red[9];
    const int tid = threadIdx.x;
    float v = 0.0f;
    for (int i = tid; i < S_TOT; i += BLK) v += ws[i];   // fixed order per thread
    const float t = block_sum(v, s_red, tid);
    if (tid == 0) out[0] = t;
}

extern "C" void kernel_launch(void* const* d_in, const int* in_sizes, int n_in,
                              void* d_out, int out_size, void* d_ws, size_t ws_size,
                              hipStream_t stream) {
    const float* alpha = (const float*)d_in[0];   // [8192,64,16,8]
    const float* beta  = (const float*)d_in[1];   // [8192,2016]
    const float* theta = (const float*)d_in[2];   // [8192]
    float* out = (float*)d_out;                   // scalar
    float* ws  = (float*)d_ws;                    // >= 8192 floats

    hipLaunchKernelGGL(depth_kernel, dim3(S_TOT), dim3(BLK), 0, stream,
                       alpha, beta, theta, ws);
    hipLaunchKernelGGL(final_reduce_kernel, dim3(1), dim3(BLK), 0, stream,
                       ws, out);
}